// EdgePoint2Wrapper_32744830665317
// MI455X (gfx1250) — compile-verified
//
#include <hip/hip_runtime.h>
#include <hip/hip_bf16.h>

#define BATCH 4
#define CHANS 64
#define HH 768
#define WW 768
#define NPIX (HH * WW)          // 589824
#define TOPK 2048
#define TILE 16
#define HALO 2
#define TDIM (TILE + 2 * HALO)  // 20
#define CAP  131072             // per-batch candidate capacity (~5x expected)

typedef unsigned int u32x4 __attribute__((ext_vector_type(4)));
typedef int i32x4 __attribute__((ext_vector_type(4)));
typedef int i32x8 __attribute__((ext_vector_type(8)));

static __device__ __forceinline__ unsigned f32_to_key(float f) {
  unsigned u = __float_as_uint(f);
  return (u & 0x80000000u) ? ~u : (u | 0x80000000u);
}

static __device__ __forceinline__ void wait_async0() {
#if __has_builtin(__builtin_amdgcn_s_wait_asynccnt)
  __builtin_amdgcn_s_wait_asynccnt(0);
#else
  asm volatile("s_wait_asynccnt 0" ::: "memory");
#endif
}

static __device__ __forceinline__ void wait_tensor0() {
#if __has_builtin(__builtin_amdgcn_s_wait_tensorcnt)
  __builtin_amdgcn_s_wait_tensorcnt(0);
#else
  asm volatile("s_wait_tensorcnt 0" ::: "memory");
#endif
}

// meta layout per batch (8 uints):
// [0]=thresholdKey [1]=countGreater [2]=tieNeed [3]=gtCtr [4]=tieCtr [5]=candCnt
__global__ __launch_bounds__(32) void k_init(unsigned* __restrict__ meta) {
  if (threadIdx.x < BATCH * 8) meta[threadIdx.x] = 0u;
}

// ---------------------------------------------------------------------------
// K1: 5x5 NMS + threshold + border; halo tile staged into LDS by the Tensor
// Data Mover (one 2D-tile D# per workgroup, TENSORcnt-tracked). Survivors are
// compacted into a per-batch candidate list (wave ballot, 1 atomic per wave).
// ---------------------------------------------------------------------------
__global__ __launch_bounds__(256) void k_nms_cand(const float* __restrict__ det,
                                                  unsigned long long* __restrict__ cand,
                                                  unsigned* __restrict__ meta) {
  __shared__ __align__(16) float tile[TDIM * TDIM];
  const int tx = threadIdx.x, ty = threadIdx.y;
  const int b = blockIdx.z;
  const int t = ty * TILE + tx;
  // Clamp tile origin into the tensor; border mask guarantees every surviving
  // output pixel's 5x5 window stays inside the clamped 20x20 tile.
  const int x0c = min(max((int)blockIdx.x * TILE - HALO, 0), WW - TDIM);
  const int y0c = min(max((int)blockIdx.y * TILE - HALO, 0), HH - TDIM);
  const float* p0 = det + ((size_t)b * NPIX + (size_t)y0c * WW + x0c);

#if __has_builtin(__builtin_amdgcn_tensor_load_to_lds)
  if (t < 32) {  // one TDM issue per workgroup (wave 0)
    const unsigned lds_base = (unsigned)(unsigned long long)(uintptr_t)&tile[0];
    const unsigned long long ga = (unsigned long long)(uintptr_t)p0;
    // D# group0: count=1 | lds_addr | global_addr[56:0] | type=2
    u32x4 g0 = {1u, lds_base, (unsigned)ga,
                ((unsigned)((ga >> 32) & 0x01FFFFFFu)) | 0x80000000u};
    // D# group1: data_size=4B; tensor 768x768; tile 20x20; dim0 stride 768
    i32x8 g1 = {(int)0x00020000,        // data_size=2 (4B)
                (int)0x03000000,        // tensor_dim0=768 -> bits[79:48] lo
                (int)0x03000000,        // tensor_dim1=768 -> bits[111:80] lo
                (int)(TDIM << 16),      // tile_dim0=20 -> bits[127:112]
                (int)TDIM,              // tile_dim1=20 (tile_dim2=0)
                (int)WW,                // tensor_dim0_stride=768 (lo 32 of 48)
                0, 0};                  // stride hi / tensor_dim1_stride=0
    i32x4 gz4 = {0, 0, 0, 0};
#if __has_include(<hip/amd_detail/amd_gfx1250_TDM.h>)
    i32x8 gz8 = {0, 0, 0, 0, 0, 0, 0, 0};
    __builtin_amdgcn_tensor_load_to_lds(g0, g1, gz4, gz4, gz8, 0);
#else
    __builtin_amdgcn_tensor_load_to_lds(g0, g1, gz4, gz4, 0);
#endif
    wait_tensor0();
  }
#else
  // Fallback: per-lane async copy engine (ASYNCcnt-tracked).
  for (int l = t; l < TDIM * TDIM; l += 256) {
    int r = l / TDIM, c = l % TDIM;
    const float* p = p0 + (size_t)r * WW + c;
    unsigned loff = (unsigned)(unsigned long long)(uintptr_t)&tile[l];
    asm volatile("global_load_async_to_lds_b32 %0, %1, off"
                 :: "v"(loff), "v"(p) : "memory");
  }
  wait_async0();
#endif
  __syncthreads();

  const int gx = blockIdx.x * TILE + tx;
  const int gy = blockIdx.y * TILE + ty;
  const bool border = (gx >= 4) & (gx < WW - 4) & (gy >= 4) & (gy < HH - 4);

  bool keep = false;
  unsigned key = 0u;
  if (border) {
    const int lx = gx - x0c;            // in [2,17] when border holds
    const int ly = gy - y0c;
    const float c = tile[ly * TDIM + lx];
    float m = c;
#pragma unroll
    for (int dy = -2; dy <= 2; ++dy)
#pragma unroll
      for (int dx = -2; dx <= 2; ++dx)
        m = fmaxf(m, tile[(ly + dy) * TDIM + (lx + dx)]);
    keep = (c == m) & (c > 0.0f);
    key = f32_to_key(c);                // > 0 => key >= 0x80000000
  }

  const unsigned mask = (unsigned)__ballot(keep);
  if (mask) {
    const unsigned lane = (unsigned)(t & 31);
    const int leader = __ffs(mask) - 1;
    unsigned base = 0u;
    if ((int)lane == leader) base = atomicAdd(&meta[b * 8 + 5], __popc(mask));
    base = __shfl(base, leader, 32);
    if (keep) {
      const unsigned pos = base + __popc(mask & ((1u << lane) - 1u));
      if (pos < CAP) {
        const unsigned idx = (unsigned)(gy * WW + gx);
        cand[(size_t)b * CAP + pos] =
            ((unsigned long long)key << 32) | (unsigned)(~idx);
      }
    }
  }
}

// ---------------------------------------------------------------------------
// K2: exact 2048th-largest key over the candidate list (4x8-bit MSB radix,
// one block per batch, LDS histograms). If fewer than TOPK candidates, emits
// -inf filler entries in index order.
// ---------------------------------------------------------------------------
__global__ __launch_bounds__(1024) void k_select(const unsigned long long* __restrict__ cand,
                                                 unsigned* __restrict__ meta,
                                                 unsigned long long* __restrict__ sel) {
  __shared__ unsigned hist[256];
  __shared__ unsigned sh_prefix, sh_needed, sh_greater;
  const int b = blockIdx.x;
  const unsigned t = threadIdx.x;
  const unsigned n = min(meta[b * 8 + 5], (unsigned)CAP);
  const unsigned long long* cb = cand + (size_t)b * CAP;

  if (n >= TOPK) {
    if (t == 0) { sh_prefix = 0u; sh_needed = TOPK; sh_greater = 0u; }
    for (int pass = 0; pass < 4; ++pass) {
      const int shift = 24 - 8 * pass;
      if (t < 256) hist[t] = 0u;
      __syncthreads();
      const unsigned pref = sh_prefix;
      for (unsigned i = t; i < n; i += 1024) {
        const unsigned key = (unsigned)(cb[i] >> 32);
        if (pass == 0 || (key >> (shift + 8)) == pref)
          atomicAdd(&hist[(key >> shift) & 0xFFu], 1u);
      }
      __syncthreads();
      if (t == 0) {
        unsigned needed = sh_needed, cum = 0u;
        int bin = 255;
        for (; bin > 0; --bin) {
          const unsigned h = hist[bin];
          if (cum + h >= needed) break;
          cum += h;
        }
        sh_greater += cum;
        sh_needed = needed - cum;
        sh_prefix = (pref << 8) | (unsigned)bin;
      }
      __syncthreads();
    }
    if (t == 0) {
      unsigned* m = meta + b * 8;
      m[0] = sh_prefix; m[1] = sh_greater; m[2] = sh_needed;
      m[3] = 0u; m[4] = 0u;
    }
  } else {
    if (t == 0) {
      unsigned* m = meta + b * 8;
      m[0] = 0u; m[1] = n; m[2] = 0u; m[3] = 0u; m[4] = 0u;
    }
    for (unsigned r = t; r + n < TOPK; r += 1024)      // -inf fillers, idx asc
      sel[(size_t)b * TOPK + n + r] =
          (0x007FFFFFull << 32) | (unsigned)(~r);
  }
}

// ---------------------------------------------------------------------------
// K3: compaction of the winning 2048 into sel (slots disjoint & exact).
// ---------------------------------------------------------------------------
__global__ __launch_bounds__(256) void k_compact(const unsigned long long* __restrict__ cand,
                                                 unsigned* __restrict__ meta,
                                                 unsigned long long* __restrict__ sel) {
  const int b = blockIdx.y;
  const unsigned i = blockIdx.x * 256 + threadIdx.x;
  unsigned* m = meta + b * 8;
  const unsigned n = min(m[5], (unsigned)CAP);
  if (i >= n) return;
  const unsigned long long pk = cand[(size_t)b * CAP + i];
  const unsigned key = (unsigned)(pk >> 32);
  const unsigned T = m[0], G = m[1], tieNeed = m[2];
  if (key > T) {
    const unsigned pos = atomicAdd(&m[3], 1u);
    sel[(size_t)b * TOPK + pos] = pk;
  } else if (key == T) {
    const unsigned tp = atomicAdd(&m[4], 1u);
    if (tp < tieNeed) sel[(size_t)b * TOPK + G + tp] = pk;
  }
}

// ---------------------------------------------------------------------------
// K4: async-b128 LDS fill + bitonic sort of 2048 u64 composites (descending:
// score desc, index asc on ties), then emit kpts/scores/valid + sorted list.
// ---------------------------------------------------------------------------
__global__ __launch_bounds__(1024) void k_sort_emit(unsigned long long* __restrict__ sel,
                                                    float* __restrict__ out) {
  __shared__ __align__(16) unsigned long long s[TOPK];
  const int b = blockIdx.x;
  const unsigned t = threadIdx.x;
  {
    const unsigned long long* gp = sel + (size_t)b * TOPK + 2u * t;
    unsigned loff = (unsigned)(unsigned long long)(uintptr_t)&s[2u * t];
    asm volatile("global_load_async_to_lds_b128 %0, %1, off"
                 :: "v"(loff), "v"(gp) : "memory");
  }
  wait_async0();
  __syncthreads();

  for (unsigned k = 2; k <= TOPK; k <<= 1) {
    for (unsigned j = k >> 1; j > 0; j >>= 1) {
      __syncthreads();
      for (unsigned i = t; i < TOPK; i += 1024) {
        const unsigned l = i ^ j;
        if (l > i) {
          const unsigned long long a = s[i], c = s[l];
          const bool swp = ((i & k) == 0) ? (a < c) : (a > c);  // descending
          if (swp) { s[i] = c; s[l] = a; }
        }
      }
    }
  }
  __syncthreads();

  float* kpts   = out;                                   // (B,K,2)
  float* scores = out + (size_t)BATCH * TOPK * 2;        // (B,K)
  float* validf = out + (size_t)BATCH * TOPK * 2 +
                  (size_t)BATCH * TOPK +
                  (size_t)BATCH * TOPK * CHANS;          // (B,K)
  for (unsigned i = t; i < TOPK; i += 1024) {
    const unsigned long long pr = s[i];
    const unsigned key = (unsigned)(pr >> 32);
    const unsigned idx = ~((unsigned)pr);
    const bool vld = key >= 0x80000000u;                 // finite positive score
    const float val = vld ? __uint_as_float(key & 0x7FFFFFFFu) : 0.0f;
    const size_t o = (size_t)b * TOPK + i;
    kpts[o * 2 + 0] = (float)(idx % WW);
    kpts[o * 2 + 1] = (float)(idx / WW);
    scores[o] = val;
    validf[o] = vld ? 1.0f : 0.0f;
    sel[o] = pr;                                         // sorted, for K5
  }
}

// ---------------------------------------------------------------------------
// K5: bilinear descriptor sampling (lane == channel), exact reference math.
// ---------------------------------------------------------------------------
__global__ __launch_bounds__(64) void k_sample(const float* __restrict__ desc,
                                               const unsigned long long* __restrict__ sel,
                                               float* __restrict__ out) {
  const int k = blockIdx.x, b = blockIdx.y, c = threadIdx.x;
  float* dst = out + (size_t)BATCH * TOPK * 2 + (size_t)BATCH * TOPK +
               ((size_t)b * TOPK + k) * CHANS;
  const unsigned long long pr = sel[(size_t)b * TOPK + k];
  const unsigned key = (unsigned)(pr >> 32);
  if (key < 0x80000000u) { dst[c] = 0.0f; return; }
  const unsigned idx = ~((unsigned)pr);
  const float xf = (float)(idx % WW);
  const float yf = (float)(idx / WW);

  const float gx = (xf + 0.5f) / (float)WW * 2.0f - 1.0f;
  const float gy = (yf + 0.5f) / (float)HH * 2.0f - 1.0f;
  const float ix = (gx + 1.0f) * ((float)WW * 0.5f) - 0.5f;
  const float iy = (gy + 1.0f) * ((float)HH * 0.5f) - 0.5f;
  const float x0 = floorf(ix), y0 = floorf(iy);
  const float wx = ix - x0, wy = iy - y0;
  const int x0i = (int)x0, y0i = (int)y0;

  const float* plane = desc + ((size_t)b * CHANS + c) * (size_t)NPIX;
  auto g = [&](int xi, int yi) -> float {
    const bool inb = (xi >= 0) & (xi < WW) & (yi >= 0) & (yi < HH);
    const int xc = min(max(xi, 0), WW - 1);
    const int yc = min(max(yi, 0), HH - 1);
    const float v = plane[(size_t)yc * WW + xc];
    return inb ? v : 0.0f;
  };
  const float v00 = g(x0i,     y0i);
  const float v01 = g(x0i + 1, y0i);
  const float v10 = g(x0i,     y0i + 1);
  const float v11 = g(x0i + 1, y0i + 1);
  dst[c] = v00 * (1.0f - wx) * (1.0f - wy) + v01 * wx * (1.0f - wy) +
           v10 * (1.0f - wx) * wy + v11 * wx * wy;
}

// ---------------------------------------------------------------------------
extern "C" void kernel_launch(void* const* d_in, const int* in_sizes, int n_in,
                              void* d_out, int out_size, void* d_ws, size_t ws_size,
                              hipStream_t stream) {
  const float* raw_desc   = (const float*)d_in[0];  // (4,64,768,768)
  const float* raw_detect = (const float*)d_in[1];  // (4,1,768,768)
  (void)in_sizes; (void)n_in; (void)out_size; (void)ws_size;

  const size_t CAND_BYTES = (size_t)BATCH * CAP * sizeof(unsigned long long); // 4 MB
  const size_t SEL_BYTES  = (size_t)BATCH * TOPK * sizeof(unsigned long long);
  unsigned long long* cand = (unsigned long long*)d_ws;
  unsigned long long* sel  = (unsigned long long*)((char*)d_ws + CAND_BYTES);
  unsigned* meta = (unsigned*)((char*)d_ws + CAND_BYTES + SEL_BYTES);
  float* out = (float*)d_out;

  k_init<<<dim3(1), dim3(32), 0, stream>>>(meta);

  dim3 g1(WW / TILE, HH / TILE, BATCH), b1(TILE, TILE);
  k_nms_cand<<<g1, b1, 0, stream>>>(raw_detect, cand, meta);

  k_select<<<dim3(BATCH), dim3(1024), 0, stream>>>(cand, meta, sel);

  k_compact<<<dim3(CAP / 256, BATCH), dim3(256), 0, stream>>>(cand, meta, sel);

  k_sort_emit<<<dim3(BATCH), dim3(1024), 0, stream>>>(sel, out);

  k_sample<<<dim3(TOPK, BATCH), dim3(64), 0, stream>>>(raw_desc, sel, out);
}